// DigitCapsuleLayer_29557964931531
// MI455X (gfx1250) — compile-verified
//
#include <hip/hip_runtime.h>
#include <math.h>

typedef __attribute__((ext_vector_type(2))) float v2f;
typedef __attribute__((ext_vector_type(8))) float v8f;

#define B_TOT 128
#define P_TOT 1152
#define IN_DIM 8
#define O_TOT 10
#define D_TOT 16
#define ND 160                 // O_TOT * D_TOT
#define BT 16                  // batch tile (WMMA M)
#define WAVES 4
#define PPW 18                 // p's per wave
#define PCHUNK (WAVES * PPW)   // 72 -> 16 chunks cover P=1152
#define TILE_ELEMS (BT * ND)   // 2560 floats per u_hat / s tile
#define WAVE_LDS (TILE_ELEMS * 2 + ND + 32)   // uhat + sacc + c + rowmax/rowsum = 5312
#define LDS_FLOATS (TILE_ELEMS + WAVES * WAVE_LDS) // 2560 + 21248 = 23808

// One fused routing pass: recompute u_hat tiles with V_WMMA_F32_16X16X4_F32,
// logits = u_hat . vsum, softmax over o, accumulate s = sum_p c * u_hat.
__global__ __launch_bounds__(WAVES * 32)
void routing_pass(const float* __restrict__ x, const float* __restrict__ W,
                  const float* __restrict__ vsum_g, float* __restrict__ s_g)
{
    extern __shared__ float lds[];
    const int tid  = threadIdx.x;
    const int wave = tid >> 5;
    const int lane = tid & 31;

    float* vsum_l = lds;                                   // 2560, shared
    float* uh  = lds + TILE_ELEMS + wave * WAVE_LDS;       // 2560 per wave
    float* sa  = uh + TILE_ELEMS;                          // 2560 per wave
    float* cc  = sa + TILE_ELEMS;                          // 160 per wave
    float* rmx = cc + ND;                                  // 16
    float* rsm = rmx + 16;                                 // 16

    const int b0 = blockIdx.x * BT;

    // Stage vsum tile (read-only this pass) and zero per-wave s accumulator.
    for (int i = tid; i < TILE_ELEMS; i += blockDim.x)
        vsum_l[i] = vsum_g[(size_t)b0 * ND + i];
#pragma unroll
    for (int k = 0; k < TILE_ELEMS / 32; ++k)
        sa[lane + 32 * k] = 0.0f;
    __syncthreads();

    const int h  = lane >> 4;   // lane half selects K-half / M-half
    const int lm = lane & 15;
    const int pbase = blockIdx.y * PCHUNK + wave * PPW;

    for (int pp = 0; pp < PPW; ++pp) {
        const int p = pbase + pp;

        // A operand: x[b0+lm, p, i]; lane half h covers K = {2h,2h+1} (lo) / {4+2h,4+2h+1} (hi)
        const float* xp = x + ((size_t)(b0 + lm) * P_TOT + p) * IN_DIM;
        v2f alo = { xp[2 * h + 0], xp[2 * h + 1] };
        v2f ahi = { xp[4 + 2 * h + 0], xp[4 + 2 * h + 1] };

#pragma unroll
        for (int o = 0; o < O_TOT; ++o) {
            // B operand: W[p,o,d,i] with K=i, N=d=lm
            const float* wp = W + (((size_t)p * O_TOT + o) * D_TOT + lm) * IN_DIM;
            v2f blo = { wp[2 * h + 0], wp[2 * h + 1] };
            v2f bhi = { wp[4 + 2 * h + 0], wp[4 + 2 * h + 1] };

            v8f acc = {};
            acc = __builtin_amdgcn_wmma_f32_16x16x4_f32(
                      false, alo, false, blo, (short)0, acc, false, false);
            acc = __builtin_amdgcn_wmma_f32_16x16x4_f32(
                      false, ahi, false, bhi, (short)0, acc, false, false);

            // C layout: vgpr r, lane -> (M = r + 8h, N = lm)
#pragma unroll
            for (int r = 0; r < 8; ++r)
                uh[(r + 8 * h) * ND + o * D_TOT + lm] = acc[r];
        }

        // logits[row,o] = u_hat[row,o,:] . vsum[row,o,:]  (160 items over 32 lanes)
#pragma unroll
        for (int k = 0; k < 5; ++k) {
            const int idx = lane + 32 * k;      // 0..159
            const int row = idx & 15;
            const int o   = idx >> 4;
            float dot = 0.0f;
#pragma unroll
            for (int d = 0; d < D_TOT; ++d)
                dot += uh[row * ND + o * D_TOT + d] * vsum_l[row * ND + o * D_TOT + d];
            cc[o * 16 + row] = dot;
        }

        // per-row softmax normalizers (lanes 0..15, one row each)
        if (lane < 16) {
            float m = -3.4e38f;
            for (int o = 0; o < O_TOT; ++o) m = fmaxf(m, cc[o * 16 + lane]);
            float sum = 0.0f;
            for (int o = 0; o < O_TOT; ++o) sum += __expf(cc[o * 16 + lane] - m);
            rmx[lane] = m;
            rsm[lane] = sum;
        }

        // convert logits -> c in place (each lane owns its elements)
#pragma unroll
        for (int k = 0; k < 5; ++k) {
            const int idx = lane + 32 * k;
            const int row = idx & 15;
            const int o   = idx >> 4;
            cc[o * 16 + row] = __expf(cc[o * 16 + row] - rmx[row]) / rsm[row];
        }

        // s[row, o*16+d] += c[row,o] * u_hat[row, o*16+d]
#pragma unroll
        for (int k = 0; k < TILE_ELEMS / 32; ++k) {
            const int e   = lane + 32 * k;
            const int row = e / ND;
            const int n   = e % ND;
            sa[e] += cc[(n >> 4) * 16 + row] * uh[e];
        }
    }

    __syncthreads();
    // Combine the 4 wave-local accumulators, one global atomic per element.
    for (int i = tid; i < TILE_ELEMS; i += blockDim.x) {
        float acc = 0.0f;
#pragma unroll
        for (int w = 0; w < WAVES; ++w)
            acc += lds[TILE_ELEMS + w * WAVE_LDS + TILE_ELEMS + i];
        atomicAdd(&s_g[(size_t)b0 * ND + i], acc);
    }
}

// v = squash(s); either accumulate into vsum (mid iterations) or emit output.
__global__ void squash_kernel(const float* __restrict__ s_g,
                              float* __restrict__ vsum_g,
                              float* __restrict__ out_g)
{
    const int t = blockIdx.x * blockDim.x + threadIdx.x;
    if (t >= B_TOT * O_TOT) return;
    const int b = t / O_TOT;
    const int o = t % O_TOT;
    const float* sp = s_g + (size_t)b * ND + o * D_TOT;

    float sv[D_TOT];
    float nrm2 = 0.0f;
#pragma unroll
    for (int d = 0; d < D_TOT; ++d) { sv[d] = sp[d]; nrm2 += sv[d] * sv[d]; }
    const float nrm   = sqrtf(nrm2);
    const float scale = (nrm2 / (1.0f + nrm2)) / (nrm + 1e-8f);

    if (vsum_g) {
        float* vp = vsum_g + (size_t)b * ND + o * D_TOT;
#pragma unroll
        for (int d = 0; d < D_TOT; ++d) vp[d] += scale * sv[d];
    }
    if (out_g) {
        float* op = out_g + (size_t)b * ND + o * D_TOT;
#pragma unroll
        for (int d = 0; d < D_TOT; ++d) op[d] = scale * sv[d];
    }
}

extern "C" void kernel_launch(void* const* d_in, const int* in_sizes, int n_in,
                              void* d_out, int out_size, void* d_ws, size_t ws_size,
                              hipStream_t stream)
{
    const float* x = (const float*)d_in[0];   // [128,1152,8]
    const float* W = (const float*)d_in[1];   // [1,1152,10,16,8]

    float* s    = (float*)d_ws;               // [128,10,16]
    float* vsum = s + (size_t)B_TOT * ND;     // [128,10,16]
    const size_t bytesBOD = (size_t)B_TOT * ND * sizeof(float);

    hipMemsetAsync(vsum, 0, bytesBOD, stream);

    const dim3 grid(B_TOT / BT, P_TOT / PCHUNK);
    const size_t ldsBytes = (size_t)LDS_FLOATS * sizeof(float);

    for (int it = 0; it < 3; ++it) {
        hipMemsetAsync(s, 0, bytesBOD, stream);
        routing_pass<<<grid, WAVES * 32, ldsBytes, stream>>>(x, W, vsum, s);
        const bool last = (it == 2);
        squash_kernel<<<(B_TOT * O_TOT + 127) / 128, 128, 0, stream>>>(
            s, last ? nullptr : vsum, last ? (float*)d_out : nullptr);
    }
}